// TowerGCN_41360535060601
// MI455X (gfx1250) — compile-verified
//
#include <hip/hip_runtime.h>
#include <hip/hip_bf16.h>

// ---------------------------------------------------------------------------
// TowerGCN on gfx1250 (MI455X): 2x GCNConv (D=64) + window-64 avg pool.
// GEMMs run on V_WMMA_F32_16X16X4_F32 (fp32 matrix pipe, exact precision);
// edge aggregation uses L2-resident f32 global atomics (all buffers << 192MB L2).
// ---------------------------------------------------------------------------

typedef float v2f __attribute__((ext_vector_type(2)));
typedef float v8f __attribute__((ext_vector_type(8)));

#define N_NODES 65536
#define N_EDGES 1048576
#define DIM 64
#define M_POOL (N_NODES / DIM)   // 1024 pooling windows

// ---------------- degree / norm prep ----------------

__global__ __launch_bounds__(256) void k_init_deg(float* __restrict__ deg) {
    int i = blockIdx.x * 256 + threadIdx.x;
    deg[i] = 1.0f;  // self-loop weight
}

__global__ __launch_bounds__(256) void k_deg_add(const int* __restrict__ col,
                                                 const float* __restrict__ w,
                                                 float* __restrict__ deg) {
    int e = blockIdx.x * 256 + threadIdx.x;
    atomicAdd(&deg[col[e]], w[e]);
}

__global__ __launch_bounds__(256) void k_rsqrt_inplace(float* __restrict__ d) {
    int i = blockIdx.x * 256 + threadIdx.x;
    d[i] = rsqrtf(d[i]);  // deg >= 1 always (self loop), so no zero guard needed
}

// ---------------- WMMA GEMM: C[N_NODES x 64] = A[N_NODES x 64] * B[64 x 64] ----------------
// One wave computes a 16-row strip across all 64 output columns (4 16x16 tiles).
// Fragment layouts per CDNA5 ISA §7.12.2 (32-bit A 16x4, B 4x16, C/D 16x16).

__global__ __launch_bounds__(256) void k_gemm_wmma(const float* __restrict__ A,
                                                   const float* __restrict__ B,
                                                   float* __restrict__ C) {
    const int gwave = (blockIdx.x * 256 + (int)threadIdx.x) >> 5;
    const int lane  = threadIdx.x & 31;
    const int half  = lane >> 4;   // 0: lanes 0-15, 1: lanes 16-31
    const int r     = lane & 15;
    const int m0    = gwave << 4;  // 16 rows per wave

    v8f acc0 = {}, acc1 = {}, acc2 = {}, acc3 = {};

    const float* arow = A + (size_t)(m0 + r) * DIM + 2 * half;

#pragma unroll
    for (int k0 = 0; k0 < DIM; k0 += 4) {
        // A fragment: lane (half,r) holds A[m0+r][k0+2*half .. +1]
        v2f a;
        a.x = arow[k0];
        a.y = arow[k0 + 1];

        const float* b0 = B + (size_t)(k0 + 2 * half) * DIM + r;
        v2f b;

        b.x = b0[0];        b.y = b0[DIM];        // N-tile 0: cols 0..15
        acc0 = __builtin_amdgcn_wmma_f32_16x16x4_f32(false, a, false, b, (short)0, acc0, false, false);
        b.x = b0[16];       b.y = b0[DIM + 16];   // N-tile 1: cols 16..31
        acc1 = __builtin_amdgcn_wmma_f32_16x16x4_f32(false, a, false, b, (short)0, acc1, false, false);
        b.x = b0[32];       b.y = b0[DIM + 32];   // N-tile 2: cols 32..47
        acc2 = __builtin_amdgcn_wmma_f32_16x16x4_f32(false, a, false, b, (short)0, acc2, false, false);
        b.x = b0[48];       b.y = b0[DIM + 48];   // N-tile 3: cols 48..63
        acc3 = __builtin_amdgcn_wmma_f32_16x16x4_f32(false, a, false, b, (short)0, acc3, false, false);
    }

    // C/D layout: VGPR v, lane(half,r) -> row m0 + v + 8*half, col r (+16*n)
    float* crow = C + (size_t)(m0 + 8 * half) * DIM + r;
#pragma unroll
    for (int v = 0; v < 8; ++v) {
        crow[(size_t)v * DIM + 0]  = acc0[v];
        crow[(size_t)v * DIM + 16] = acc1[v];
        crow[(size_t)v * DIM + 32] = acc2[v];
        crow[(size_t)v * DIM + 48] = acc3[v];
    }
}

// ---------------- edge scatter aggregation ----------------

__global__ __launch_bounds__(256) void k_zero4(float4* __restrict__ p) {
    int i = blockIdx.x * 256 + threadIdx.x;
    p[i] = make_float4(0.f, 0.f, 0.f, 0.f);
}

// one wave per edge; each lane owns 2 feature dims
__global__ __launch_bounds__(256) void k_edge_agg(const int* __restrict__ row,
                                                  const int* __restrict__ col,
                                                  const float* __restrict__ w,
                                                  const float* __restrict__ dinv,
                                                  const float* __restrict__ h,
                                                  float* __restrict__ out) {
    const int t    = blockIdx.x * 256 + threadIdx.x;
    const int e    = t >> 5;
    const int lane = t & 31;
    const int rs = row[e];
    const int cs = col[e];
    const float norm = dinv[rs] * w[e] * dinv[cs];
    const float2 hv = *(const float2*)(h + (size_t)rs * DIM + lane * 2);
    float* o = out + (size_t)cs * DIM + lane * 2;
    atomicAdd(o,     norm * hv.x);
    atomicAdd(o + 1, norm * hv.y);
}

// out += h * dinv^2 + b   (self-loop message + bias), float4 vectorized
__global__ __launch_bounds__(256) void k_self_bias(const float* __restrict__ h,
                                                   const float* __restrict__ dinv,
                                                   const float* __restrict__ b,
                                                   float* __restrict__ out) {
    const int t    = blockIdx.x * 256 + threadIdx.x;  // one float4 per thread
    const int node = t >> 4;                          // 16 float4 per row
    const int d4   = (t & 15) * 4;
    const float di = dinv[node];
    const float s  = di * di;
    const float4 hv = *(const float4*)(h + (size_t)node * DIM + d4);
    const float4 bv = *(const float4*)(b + d4);
    float4 ov = *(const float4*)(out + (size_t)node * DIM + d4);
    ov.x += hv.x * s + bv.x;
    ov.y += hv.y * s + bv.y;
    ov.z += hv.z * s + bv.z;
    ov.w += hv.w * s + bv.w;
    *(float4*)(out + (size_t)node * DIM + d4) = ov;
}

// ---------------- pooling: mean over windows of 64 nodes, output [64, 1024] ----------------

__global__ __launch_bounds__(256) void k_pool(const float* __restrict__ g,
                                              float* __restrict__ out) {
    const int t = blockIdx.x * 256 + threadIdx.x;  // [0, 65536)
    const int m = t >> 6;     // window index
    const int d = t & 63;     // feature dim
    const float* base = g + (size_t)m * DIM * DIM + d;
    float s = 0.f;
#pragma unroll 8
    for (int j = 0; j < DIM; ++j) s += base[(size_t)j * DIM];
    out[(size_t)d * M_POOL + m] = s * (1.0f / DIM);
}

// ---------------- launch ----------------

extern "C" void kernel_launch(void* const* d_in, const int* in_sizes, int n_in,
                              void* d_out, int out_size, void* d_ws, size_t ws_size,
                              hipStream_t stream) {
    const float* x   = (const float*)d_in[0];
    const int*  eidx = (const int*) d_in[1];   // [2, E]
    const float* ew  = (const float*)d_in[2];
    const float* W1  = (const float*)d_in[3];
    const float* b1  = (const float*)d_in[4];
    const float* W2  = (const float*)d_in[5];
    const float* b2  = (const float*)d_in[6];
    float* out = (float*)d_out;

    const int* rowi = eidx;             // edge_index[0]
    const int* coli = eidx + N_EDGES;   // edge_index[1]

    // workspace: dinv (64K f32) | bufH (4M f32) | bufA (4M f32)  ~= 34 MB
    float* dinv = (float*)d_ws;
    float* bufH = dinv + N_NODES;
    float* bufA = bufH + (size_t)N_NODES * DIM;

    const int ZERO_BLK = (N_NODES * DIM / 4) / 256;   // 4096
    const int AGG_BLK  = (int)(((long long)N_EDGES * 32) / 256);  // 131072
    const int SB_BLK   = (N_NODES * 16) / 256;        // 4096
    const int GEMM_BLK = (N_NODES / 16) / 8;          // 512 blocks of 8 waves

    // normalization: deg = 1 + scatter(ew at col); dinv = rsqrt(deg)
    k_init_deg<<<N_NODES / 256, 256, 0, stream>>>(dinv);
    k_deg_add<<<N_EDGES / 256, 256, 0, stream>>>(coli, ew, dinv);
    k_rsqrt_inplace<<<N_NODES / 256, 256, 0, stream>>>(dinv);

    // ---- layer 1 ----
    k_gemm_wmma<<<GEMM_BLK, 256, 0, stream>>>(x, W1, bufH);
    k_zero4<<<ZERO_BLK, 256, 0, stream>>>((float4*)bufA);
    k_edge_agg<<<AGG_BLK, 256, 0, stream>>>(rowi, coli, ew, dinv, bufH, bufA);
    k_self_bias<<<SB_BLK, 256, 0, stream>>>(bufH, dinv, b1, bufA);

    // ---- layer 2 (ping-pong buffers) ----
    k_gemm_wmma<<<GEMM_BLK, 256, 0, stream>>>(bufA, W2, bufH);
    k_zero4<<<ZERO_BLK, 256, 0, stream>>>((float4*)bufA);
    k_edge_agg<<<AGG_BLK, 256, 0, stream>>>(rowi, coli, ew, dinv, bufH, bufA);
    k_self_bias<<<SB_BLK, 256, 0, stream>>>(bufH, dinv, b2, bufA);

    // ---- pooling -> [64, 1024] ----
    k_pool<<<(M_POOL * DIM) / 256, 256, 0, stream>>>(bufA, out);
}